// YibeiSynth_74156905333421
// MI455X (gfx1250) — compile-verified
//
#include <hip/hip_runtime.h>
#include <hip/hip_bf16.h>
#include <stdint.h>

// Memory-bound volumetric relabel+gather+scale kernel for MI455X (gfx1250).
// ~256 MB of traffic, ~0.017 GFLOP -> pure HBM roofline (~11 us @ 23.3 TB/s).
// Strategy: fused {mult, mask} LDS LUT (one ds_load_b64 per voxel), 128-bit
// non-temporal streaming loads/stores, gfx1250 async-to-LDS staging of the
// LUTs, global_prefetch for the next grid-stride tile. wave32-sized blocks.

typedef int   __attribute__((ext_vector_type(4))) i32x4;
typedef float __attribute__((ext_vector_type(4))) f32x4;
typedef float __attribute__((ext_vector_type(2))) f32x2;

#define N_LABELS 1000
#define LUT_PAD  1024   // power of two so we can mask indices cheaply

#if defined(__HIP_DEVICE_COMPILE__)
#if __has_builtin(__builtin_amdgcn_global_load_async_to_lds_b32)
#define VESSEL_HAVE_ASYNC_LDS 1
#endif
#endif

#define VESSEL_AS1 __attribute__((address_space(1)))
#define VESSEL_AS3 __attribute__((address_space(3)))

__global__ __launch_bounds__(256) void vessel_prune_mul_kernel(
    const int*   __restrict__ labels,
    const int*   __restrict__ keep,
    const float* __restrict__ ilut,
    const float* __restrict__ paren,
    float*       __restrict__ out,
    float*       __restrict__ maskout,   // may be nullptr
    long long nvec)                      // number of vec4 elements
{
    // Fused LUT: s_lut[l] = { multiplier, maskbit }
    //   l == 0           -> {1.0, 0.0}   (background)
    //   keep[l] == 0     -> {1.0, 0.0}   (pruned -> behaves as background)
    //   else             -> {ilut[l], 1.0}
    __shared__ f32x2 s_lut[LUT_PAD];
    __shared__ int   s_keep[LUT_PAD];
    __shared__ float s_int[LUT_PAD];

    const int tid = threadIdx.x;
    const int nt  = blockDim.x;

    // --- Stage raw LUTs into LDS (async-to-LDS on gfx1250 when available) ---
    for (int i = tid; i < LUT_PAD; i += nt) {
        const int j = (i < N_LABELS) ? i : (N_LABELS - 1);  // clamp global reads
#if VESSEL_HAVE_ASYNC_LDS
        __builtin_amdgcn_global_load_async_to_lds_b32(
            (VESSEL_AS1 int*)(keep + j), (VESSEL_AS3 int*)(s_keep + i), 0, 0);
        __builtin_amdgcn_global_load_async_to_lds_b32(
            (VESSEL_AS1 int*)(ilut + j), (VESSEL_AS3 int*)(s_int + i), 0, 0);
#else
        s_keep[i] = keep[j];
        s_int[i]  = ilut[j];
#endif
    }
#if VESSEL_HAVE_ASYNC_LDS
#if __has_builtin(__builtin_amdgcn_s_wait_asynccnt)
    __builtin_amdgcn_s_wait_asynccnt(0);
#else
    asm volatile("s_wait_asynccnt 0" ::: "memory");
#endif
#endif
    __syncthreads();

    // --- Combine into the fused {mult, mask} table ---
    for (int i = tid; i < LUT_PAD; i += nt) {
        const bool kept = (i > 0) && (i < N_LABELS) && (s_keep[i] > 0);
        f32x2 e;
        e.x = kept ? s_int[i] : 1.0f;
        e.y = kept ? 1.0f : 0.0f;
        s_lut[i] = e;
    }
    __syncthreads();

    // --- Streaming main loop: vec4 per thread-iteration, non-temporal ---
    const i32x4* __restrict__ lab4 = (const i32x4*)labels;
    const f32x4* __restrict__ par4 = (const f32x4*)paren;
    f32x4* __restrict__ out4 = (f32x4*)out;
    f32x4* __restrict__ msk4 = (f32x4*)maskout;

    const long long stride = (long long)gridDim.x * nt;
    for (long long i = (long long)blockIdx.x * nt + tid; i < nvec; i += stride) {
        if (i + stride < nvec) {   // gfx1250 global_prefetch for the next tile
            __builtin_prefetch((const void*)&lab4[i + stride], 0, 0);
            __builtin_prefetch((const void*)&par4[i + stride], 0, 0);
        }
        const i32x4 l = __builtin_nontemporal_load(&lab4[i]);
        const f32x4 p = __builtin_nontemporal_load(&par4[i]);

        const f32x2 a = s_lut[l.x & (LUT_PAD - 1)];
        const f32x2 b = s_lut[l.y & (LUT_PAD - 1)];
        const f32x2 c = s_lut[l.z & (LUT_PAD - 1)];
        const f32x2 d = s_lut[l.w & (LUT_PAD - 1)];

        f32x4 o;
        o.x = p.x * a.x; o.y = p.y * b.x; o.z = p.z * c.x; o.w = p.w * d.x;
        __builtin_nontemporal_store(o, &out4[i]);

        if (msk4) {
            f32x4 m;
            m.x = a.y; m.y = b.y; m.z = c.y; m.w = d.y;
            __builtin_nontemporal_store(m, &msk4[i]);
        }
    }
}

extern "C" void kernel_launch(void* const* d_in, const int* in_sizes, int n_in,
                              void* d_out, int out_size, void* d_ws, size_t ws_size,
                              hipStream_t stream) {
    // setup_inputs order: vessel_labels, keep_mask, intensity_lut, parenchyma
    const int*   labels = (const int*)  d_in[0];
    const int*   keep   = (const int*)  d_in[1];
    const float* ilut   = (const float*)d_in[2];
    const float* paren  = (const float*)d_in[3];

    float* out = (float*)d_out;
    const long long n = (long long)in_sizes[0];          // D^3 voxels
    // Tuple output (out_f32, mask) concatenated: mask written as 0/1 floats
    // at d_out + n if the harness allocated room for both outputs.
    float* maskout = ((long long)out_size >= 2 * n) ? (out + n) : nullptr;

    const long long nvec = n >> 2;                       // D^3 % 4 == 0
    dim3 block(256);                                     // 8 wave32 waves
    dim3 grid(2048);                                     // grid-stride, ~8 it/thread

    hipLaunchKernelGGL(vessel_prune_mul_kernel, grid, block, 0, stream,
                       labels, keep, ilut, paren, out, maskout, nvec);
}